// Uncap_40750649704571
// MI455X (gfx1250) — compile-verified
//
#include <hip/hip_runtime.h>
#include <hip/hip_bf16.h>

typedef float v4f __attribute__((ext_vector_type(4)));
typedef int   v4i __attribute__((ext_vector_type(4)));

// Address-space-qualified 128-bit pointee types for the async-LDS builtin.
typedef v4i __attribute__((address_space(1))) as1_v4i;  // global
typedef v4i __attribute__((address_space(3))) as3_v4i;  // LDS

#define B_   8
#define U_   32
#define EN_  (U_ * U_ * U_)   // 32768 tokens per batch
#define F_   256              // features
#define F4_  (F_ / 4)         // 64 float4 per row

#if defined(__gfx1250__) && \
    __has_builtin(__builtin_amdgcn_global_load_async_to_lds_b128) && \
    __has_builtin(__builtin_amdgcn_s_wait_asynccnt)
#define USE_ASYNC_LDS 1
#else
#define USE_ASYNC_LDS 0
#endif

// ---------------------------------------------------------------------------
// Kernel (hot): masked broadcast.
//   out[b,a,c,d,f] = (a<ull[b] && c<ull[b]) ? x[b, a*32+c, f] : 0
// One block per (b,a,c) row: 8192 blocks x 256 threads (8 wave32).
// Row (1 KB) is staged via async global->LDS (ASYNCcnt path), then each
// thread replays its float4 with 8 non-temporal 128-bit stores (32 d-copies).
// Placed first in the TU so its disassembly shows in the feedback snippet.
// ---------------------------------------------------------------------------
__global__ __launch_bounds__(256) void bcast_kernel(const float* __restrict__ x,
                                                    const int* __restrict__ ull,
                                                    v4f* __restrict__ out) {
    const int row = blockIdx.x;            // b*1024 + a*32 + c
    const int b   = row >> 10;
    const int a   = (row >> 5) & (U_ - 1);
    const int c   = row & (U_ - 1);
    const int u   = ull[b];
    const bool valid = (a < u) && (c < u); // uniform across the block

    const int t     = threadIdx.x;
    const int j     = t & (F4_ - 1);       // float4 index within row (0..63)
    const int dbase = t >> 6;              // 0..3

    const v4f* xrow =
        (const v4f*)(x + ((size_t)b * EN_ + (size_t)(a * U_ + c)) * F_);

    v4f v = {0.0f, 0.0f, 0.0f, 0.0f};

#if USE_ASYNC_LDS
    __shared__ v4f tile[F4_];
    if (valid && t < F4_) {
        __builtin_amdgcn_global_load_async_to_lds_b128(
            (as1_v4i*)(xrow + t),
            (as3_v4i*)&tile[t],
            /*offset=*/0, /*cpol=*/0);
    }
    __builtin_amdgcn_s_wait_asynccnt(0);
    __syncthreads();
    if (valid) v = tile[j];
#else
    if (valid) {
        __builtin_prefetch(xrow + j, 0, 3);  // global_prefetch_b8
        v = xrow[j];
    }
#endif

    // out float4 index: (row*32 + d)*64 + j
    v4f* orow = out + (size_t)row * U_ * F4_ + j;
#pragma unroll
    for (int k = 0; k < 8; ++k) {
        __builtin_nontemporal_store(v, orow + (size_t)(dbase + 4 * k) * F4_);
    }
}

// ---------------------------------------------------------------------------
// Kernel: ull[b] = round(cbrt(sum(mask[b, :])))  -- 8 blocks x 256 threads
// ---------------------------------------------------------------------------
__global__ __launch_bounds__(256) void ull_kernel(const int* __restrict__ mask,
                                                  int* __restrict__ ull) {
    const int b = blockIdx.x;
    const int t = threadIdx.x;
    const v4i* m4 = (const v4i*)(mask + (size_t)b * EN_);

    int s = 0;
#pragma unroll
    for (int i = 0; i < EN_ / 4 / 256; ++i) {   // 32 iterations, int4 loads
        v4i m = m4[t + i * 256];
        s += m.x + m.y + m.z + m.w;
    }

    __shared__ int red[256];
    red[t] = s;
    __syncthreads();
#pragma unroll
    for (int w = 128; w > 0; w >>= 1) {
        if (t < w) red[t] += red[t + w];
        __syncthreads();
    }
    if (t == 0) {
        float cr = powf((float)red[0], 1.0f / 3.0f);  // match x ** (1/3)
        ull[b] = (int)roundf(cr);
    }
}

// ---------------------------------------------------------------------------
extern "C" void kernel_launch(void* const* d_in, const int* in_sizes, int n_in,
                              void* d_out, int out_size, void* d_ws, size_t ws_size,
                              hipStream_t stream) {
    const float* x    = (const float*)d_in[0];  // [8, 32768, 256] f32
    const int*   mask = (const int*)d_in[1];    // [8, 32768] i32
    int*         ull  = (int*)d_ws;             // 8 ints of scratch
    v4f*         out  = (v4f*)d_out;            // [8,32,32,32,256] f32

    ull_kernel<<<B_, 256, 0, stream>>>(mask, ull);
    bcast_kernel<<<B_ * U_ * U_, 256, 0, stream>>>(x, ull, out);
}